// CustomLoss_38062000177852
// MI455X (gfx1250) — compile-verified
//
#include <hip/hip_runtime.h>

typedef float v2f __attribute__((ext_vector_type(2)));
typedef float v8f __attribute__((ext_vector_type(8)));

#define TAKT 700
#define B_DIM 32
#define T_DIM 1024
#define V_DIM 1024

static __device__ __forceinline__ int imax(int a, int b) { return a > b ? a : b; }

// ---------------------------------------------------------------------------
// Kernel 0: zero the scalar accumulators.
// ---------------------------------------------------------------------------
__global__ void k_init(float* ce_sum, unsigned int* count) {
    ce_sum[0] = 0.0f;
    count[0]  = 0u;
}

// ---------------------------------------------------------------------------
// Kernel 1: fused log-softmax CE + argmax.
// One wave per 16 rows. lane L: row = base + (L%16), column half = L/16
// (cols [half*512, half*512+512)).
// Pass 1: per-row max / argmax (first-occurrence ties) / gather logits[tgt].
// Pass 2: row sums of exp(x - M) accumulated with V_WMMA_F32_16X16X4_F32:
//   A (16x4 f32, 2 VGPRs): lanes 0-15 supply K=0,1 ; lanes 16-31 supply K=2,3
//   B = all ones (layout-immune) => D[m,n] += sum_k A[m,k]  (exact f32 FMA)
// D layout (16x16 f32 C/D): lanes 0-15 hold rows 0-7 in d0..d7,
// lanes 16-31 hold rows 8-15.
// ---------------------------------------------------------------------------
__global__ __launch_bounds__(256) void k_ce_argmax(
    const float* __restrict__ logits, const int* __restrict__ tgt,
    int* __restrict__ pred, float* __restrict__ ce_sum)
{
    const int lane  = threadIdx.x & 31;
    const int waveW = (blockIdx.x << 3) + (threadIdx.x >> 5);  // 8 waves/block
    const int r     = lane & 15;
    const int half  = lane >> 4;
    const int row   = (waveW << 4) + r;
    const int t     = tgt[row];

    const float* rp = logits + (size_t)row * V_DIM + half * 512;

    // ---- Pass 1: max + argmax + capture logits[tgt] -----------------------
    float m  = -__builtin_inff();
    int   am = 0;
    float xt = -__builtin_inff();
    #pragma unroll 4
    for (int c = 0; c < 512; c += 4) {
        float4 x = *(const float4*)(rp + c);
        const int cg = half * 512 + c;
        if (x.x > m) { m = x.x; am = cg;     }
        if (x.y > m) { m = x.y; am = cg + 1; }
        if (x.z > m) { m = x.z; am = cg + 2; }
        if (x.w > m) { m = x.w; am = cg + 3; }
        xt = (t == cg    ) ? x.x : xt;
        xt = (t == cg + 1) ? x.y : xt;
        xt = (t == cg + 2) ? x.z : xt;
        xt = (t == cg + 3) ? x.w : xt;
    }
    {   // combine column halves (lane r <-> lane r+16); ties -> lower index
        float om  = __shfl_xor(m,  16, 32);
        int   oam = __shfl_xor(am, 16, 32);
        float oxt = __shfl_xor(xt, 16, 32);
        if (om > m || (om == m && oam < am)) { m = om; am = oam; }
        xt = fmaxf(xt, oxt);
    }
    if (half == 0) pred[row] = am;

    // ---- Pass 2: WMMA-accumulated exp row sums (re-read hits L2) ----------
    v8f acc0 = {};
    v8f acc1 = {};
    v2f ones; ones.x = 1.0f; ones.y = 1.0f;
    for (int c = 0; c < 512; c += 4) {
        float4 x = *(const float4*)(rp + c);
        v2f a0, a1;
        a0.x = __expf(x.x - m);
        a0.y = __expf(x.y - m);
        a1.x = __expf(x.z - m);
        a1.y = __expf(x.w - m);
        acc0 = __builtin_amdgcn_wmma_f32_16x16x4_f32(
            false, a0, false, ones, (short)0, acc0, false, false);
        acc1 = __builtin_amdgcn_wmma_f32_16x16x4_f32(
            false, a1, false, ones, (short)0, acc1, false, false);
    }

    // Unwind D: on this lane, acc[v] is the row sum of row (v + 8*half).
    float partial = 0.0f;
    #pragma unroll
    for (int v = 0; v < 8; ++v) {
        const int rr = v + 8 * half;
        const float S  = acc0[v] + acc1[v];
        const float Mr = __shfl(m,  rr, 32);   // lane rr holds row rr's max
        const float Xr = __shfl(xt, rr, 32);   // and its logits[tgt]
        partial += (Mr + __logf(S) - Xr);      // -logp[tgt] = lse - x_tgt
    }
    // partial is identical across lanes 0-15 (rows 0-7) and 16-31 (rows 8-15)
    const float other = __shfl_xor(partial, 16, 32);
    if (lane == 0) atomicAdd(ce_sum, partial + other);
}

// ---------------------------------------------------------------------------
// Kernel 2: overlap penalty. One block per batch, 1024 threads.
// Offset recurrence resets whenever pred[t] != pred[t-1]; each thread walks
// back to its (typically length-1) segment start, then counts tril pairs with
// broadcast LDS reads.
// ---------------------------------------------------------------------------
__global__ __launch_bounds__(1024) void k_overlap(
    const int* __restrict__ pred, const int* __restrict__ sizes,
    unsigned int* __restrict__ count)
{
    __shared__ int p[T_DIM];
    __shared__ int sz[T_DIM];
    __shared__ int xs[T_DIM];
    __shared__ int xe[T_DIM];
    __shared__ unsigned int blkCount;

    const int b   = blockIdx.x;
    const int tid = threadIdx.x;
    if (tid == 0) blkCount = 0;

    const int pp = pred[b * T_DIM + tid];
    const int ss = sizes[b * V_DIM + pp];
    p[tid]  = pp;
    sz[tid] = ss;
    __syncthreads();

    // e_t = sz_t + o_t ;  o_t = (p_t==p_{t-1}) ? max(e_{t-1}-TAKT, 0) : 0
    int j = tid;
    while (j > 0 && p[j] == p[j - 1]) --j;
    int e = sz[j];
    for (int k = j + 1; k <= tid; ++k) e = sz[k] + imax(e - TAKT, 0);
    const int o  = e - ss;
    const int s0 = tid * TAKT + o;
    xs[tid] = s0;
    xe[tid] = s0 + ss;
    __syncthreads();

    int local = 0;
    const int mxs = s0, mxe = s0 + ss;
    for (int t2 = 0; t2 < tid; ++t2) {   // all lanes read same t2 -> broadcast
        local += (int)((pp == p[t2]) & (mxs < xe[t2]) & (mxe > xs[t2]));
    }
    #pragma unroll
    for (int off = 16; off > 0; off >>= 1) local += __shfl_xor(local, off, 32);
    if ((tid & 31) == 0) atomicAdd(&blkCount, (unsigned int)local);
    __syncthreads();
    if (tid == 0) atomicAdd(count, blkCount);
}

// ---------------------------------------------------------------------------
// Kernel 3: combine -> scalar output. ALPHA = 1.
// ---------------------------------------------------------------------------
__global__ void k_finalize(const float* ce_sum, const unsigned int* count,
                           float* out)
{
    out[0] = ce_sum[0] / (float)(B_DIM * T_DIM)
           + (float)count[0] / (float)B_DIM;
}

// ---------------------------------------------------------------------------
extern "C" void kernel_launch(void* const* d_in, const int* in_sizes, int n_in,
                              void* d_out, int out_size, void* d_ws, size_t ws_size,
                              hipStream_t stream) {
    const float* logits = (const float*)d_in[0];   // [B,T,V] f32
    const int*   tgt    = (const int*)d_in[1];     // [B,T] i32
    const int*   sizes  = (const int*)d_in[2];     // [B,V] i32
    float*       out    = (float*)d_out;           // scalar f32

    float*        ce_sum = (float*)d_ws;
    unsigned int* count  = (unsigned int*)((char*)d_ws + 16);
    int*          pred   = (int*)((char*)d_ws + 256);   // B*T ints = 128 KB

    k_init<<<1, 1, 0, stream>>>(ce_sum, count);
    // 32768 rows / 16 rows-per-wave = 2048 waves; 8 waves/block -> 256 blocks
    k_ce_argmax<<<256, 256, 0, stream>>>(logits, tgt, pred, ce_sum);
    k_overlap<<<B_DIM, T_DIM, 0, stream>>>(pred, sizes, count);
    k_finalize<<<1, 1, 0, stream>>>(ce_sum, count, out);
}